// ModelNew_25056839205335
// MI455X (gfx1250) — compile-verified
//
#include <hip/hip_runtime.h>

typedef float v2f __attribute__((ext_vector_type(2)));
typedef float v8f __attribute__((ext_vector_type(8)));

#define B_TOT   2048
#define C_CH    512
#define C2_K    256
#define HW      49                // 7*7
#define X_PER_B (C_CH * HW)       // 25088 floats per batch in x/out
#define WV_PER_B (C2_K * HW)      // 12544 floats per batch in sp_wv

__global__ __launch_bounds__(256)
void fused_spatial_channel_scale(const float* __restrict__ x,
                                 const float* __restrict__ ch_w,
                                 const float* __restrict__ sp_wv,
                                 const float* __restrict__ sp_q,
                                 float* __restrict__ out)
{
    __shared__ float lds_wv[WV_PER_B];   // [256][49] row-major, stride 49 (conflict-free)
    __shared__ float lds_q [C2_K];
    __shared__ float lds_cw[C_CH];
    __shared__ float lds_sw[64];

    const int b = blockIdx.x;
    const int t = threadIdx.x;

    // ---- stage q, ch_weight, sp_wv into LDS with coalesced b128 loads ----
    lds_q[t] = sp_q[b * C2_K + t];                       // 256 threads == C2_K
    if (t < C_CH / 4) {
        ((float4*)lds_cw)[t] =
            ((const float4*)(ch_w + (size_t)b * C_CH))[t];
    }
    const float4* wv4 = (const float4*)(sp_wv + (size_t)b * WV_PER_B);
    for (int i = t; i < WV_PER_B / 4; i += 256)
        ((float4*)lds_wv)[i] = wv4[i];

    __syncthreads();

    // ---- phase 1: sw[hw] = sigmoid(sum_k q[k]*wv[k][hw]) via V_WMMA_F32_16X16X4_F32 ----
    // A[m,k'] = q[kb+k']   (broadcast over m; k' = vgpr + 2*(lane>=16))
    // B[k',n] = wv[kb+k'][16*wave + n]
    // D[m,n]  = partial sum for hw = 16*wave + n   (read row m=0: acc[0], lanes 0..15)
    const int wave = t >> 5;
    const int lane = t & 31;
    if (wave < 4) {
        const int col  = lane & 15;
        const int half = lane >> 4;
        const int hw   = (wave << 4) + col;
        const int hwc  = hw > (HW - 1) ? (HW - 1) : hw;   // branchless clamp: EXEC stays full
        const float* wq = lds_q  + 2 * half;
        const float* wb = lds_wv + (2 * half) * HW + hwc;

        v8f c0 = {0.f, 0.f, 0.f, 0.f, 0.f, 0.f, 0.f, 0.f};
        v8f c1 = c0;                                      // two chains hide WMMA latency
        #pragma unroll 4
        for (int k = 0; k < C2_K; k += 8) {
            v2f a0, a1, b0, b1;
            a0[0] = wq[k];            a0[1] = wq[k + 1];
            b0[0] = wb[k * HW];       b0[1] = wb[k * HW + HW];
            a1[0] = wq[k + 4];        a1[1] = wq[k + 5];
            b1[0] = wb[(k + 4) * HW]; b1[1] = wb[(k + 4) * HW + HW];
            c0 = __builtin_amdgcn_wmma_f32_16x16x4_f32(false, a0, false, b0,
                                                       (short)0, c0, false, false);
            c1 = __builtin_amdgcn_wmma_f32_16x16x4_f32(false, a1, false, b1,
                                                       (short)0, c1, false, false);
        }
        float s = c0[0] + c1[0];
        if (half == 0 && hw < HW) {
            // sigmoid via fast v_exp + single v_rcp (avoids IEEE divide expansion)
            lds_sw[hw] = __builtin_amdgcn_rcpf(1.0f + __expf(-s));
        }
    }

    __syncthreads();

    // ---- phase 2: out = x * (cw[c] + sw[hw]), float4 vectorized, coalesced ----
    const float4* x4 = (const float4*)(x   + (size_t)b * X_PER_B);
    float4*       o4 = (float4*)      (out + (size_t)b * X_PER_B);
    for (int i = t; i < X_PER_B / 4; i += 256) {
        float4 v = x4[i];
        unsigned e = 4u * (unsigned)i;
        unsigned ca = (e + 0u) / HW, ha = (e + 0u) - ca * HW;
        unsigned cb = (e + 1u) / HW, hb = (e + 1u) - cb * HW;
        unsigned cc = (e + 2u) / HW, hc = (e + 2u) - cc * HW;
        unsigned cd = (e + 3u) / HW, hd = (e + 3u) - cd * HW;
        float4 r;
        r.x = v.x * (lds_cw[ca] + lds_sw[ha]);
        r.y = v.y * (lds_cw[cb] + lds_sw[hb]);
        r.z = v.z * (lds_cw[cc] + lds_sw[hc]);
        r.w = v.w * (lds_cw[cd] + lds_sw[hd]);
        o4[i] = r;
    }
}

extern "C" void kernel_launch(void* const* d_in, const int* in_sizes, int n_in,
                              void* d_out, int out_size, void* d_ws, size_t ws_size,
                              hipStream_t stream) {
    const float* x    = (const float*)d_in[0];  // [B, 512, 7, 7]
    const float* cw   = (const float*)d_in[1];  // [B, 512, 1, 1]
    const float* spwv = (const float*)d_in[2];  // [B, 256, 7, 7]
    const float* spq  = (const float*)d_in[3];  // [B, 256, 1, 1]
    float* out = (float*)d_out;

    const int B = in_sizes[0] / X_PER_B;        // 2048
    fused_spatial_channel_scale<<<B, 256, 0, stream>>>(x, cw, spwv, spq, out);
}